// GNNStack_79800492360110
// MI455X (gfx1250) — compile-verified
//
#include <hip/hip_runtime.h>
#include <math.h>

// ---------------------------------------------------------------------------
// GAT stack for gfx1250 (MI455X): fp32 WMMA GEMMs + L2-resident edge phase.
// ---------------------------------------------------------------------------

typedef __attribute__((ext_vector_type(2))) float v2f;
typedef __attribute__((ext_vector_type(8))) float v8f;

#define FDIM 64
#define NEG_SLOPE 0.2f

// ---- order-preserving float<->uint map for atomic segment-max -------------
__device__ __forceinline__ unsigned f2ord(float f) {
    unsigned u = __float_as_uint(f);
    return (u & 0x80000000u) ? ~u : (u | 0x80000000u);
}
__device__ __forceinline__ float ord2f(unsigned u) {
    return (u & 0x80000000u) ? __uint_as_float(u & 0x7fffffffu)
                             : __uint_as_float(~u);
}

// ---------------------------------------------------------------------------
// C[M x 64] = act(X)[M x 64] @ W^T[64 x 64] + bias   via V_WMMA_F32_16X16X4_F32
// One wave computes one 16x16 tile; 4 waves per block cover N=64.
// A 16x4 layout : lanes 0-15 rows, v0={K0|K2}, v1={K1|K3}
// B 4x16 layout : lanes hold columns, v0={K0|K2}, v1={K1|K3}
// C/D layout    : vgpr r -> (M=r | M=r+8), N = lane&15
// ---------------------------------------------------------------------------
template <bool RELU_IN>
__global__ __launch_bounds__(128) void k_gemm64(const float* __restrict__ X,
                                                const float* __restrict__ W,
                                                const float* __restrict__ bias,
                                                float* __restrict__ C, int M) {
    const int lane = threadIdx.x & 31;
    const int wave = threadIdx.x >> 5;     // col tile 0..3
    const int row0 = blockIdx.x * 16;
    const int col0 = wave * 16;
    const int hl   = lane & 15;
    const int kh   = (lane >> 4) * 2;      // 0 or 2 (upper half-wave holds K+2)

    const float* __restrict__ xrow = X + (size_t)(row0 + hl) * FDIM + kh;
    const float* __restrict__ wrow = W + (size_t)(col0 + hl) * FDIM + kh;

    v8f acc = {};
#pragma unroll
    for (int k0 = 0; k0 < FDIM; k0 += 4) {
        v2f a, b;
        a.x = xrow[k0];
        a.y = xrow[k0 + 1];
        if (RELU_IN) { a.x = fmaxf(a.x, 0.f); a.y = fmaxf(a.y, 0.f); }
        b.x = wrow[k0];
        b.y = wrow[k0 + 1];
        acc = __builtin_amdgcn_wmma_f32_16x16x4_f32(false, a, false, b,
                                                    (short)0, acc, false, false);
    }

    const float bcol  = bias[col0 + hl];
    const int   rbase = row0 + ((lane >> 4) * 8);
#pragma unroll
    for (int r = 0; r < 8; ++r)
        C[(size_t)(rbase + r) * FDIM + col0 + hl] = acc[r] + bcol;
}

// ---- per-node attention partials: aI = h . att[:64], aJ = h . att[64:] ----
__global__ __launch_bounds__(256) void k_att(const float* __restrict__ H,
                                             const float* __restrict__ att,
                                             float* __restrict__ aI,
                                             float* __restrict__ aJ, int M) {
    int n = blockIdx.x * blockDim.x + threadIdx.x;
    if (n >= M) return;
    const float* h = H + (size_t)n * FDIM;
    float si = 0.f, sj = 0.f;
#pragma unroll
    for (int k = 0; k < FDIM; ++k) {
        float v = h[k];
        si = fmaf(v, att[k], si);
        sj = fmaf(v, att[FDIM + k], sj);
    }
    aI[n] = si;
    aJ[n] = sj;
}

// ---- init: agg[n][f] = bias[f] --------------------------------------------
__global__ __launch_bounds__(256) void k_init_agg(float* __restrict__ agg,
                                                  const float* __restrict__ bias,
                                                  int total) {
    int i = blockIdx.x * blockDim.x + threadIdx.x;
    if (i < total) agg[i] = bias[i & (FDIM - 1)];
}

// ---- init per-node softmax scratch ----------------------------------------
__global__ __launch_bounds__(256) void k_init_nd(unsigned* __restrict__ amaxU,
                                                 float* __restrict__ denom, int M) {
    int n = blockIdx.x * blockDim.x + threadIdx.x;
    if (n < M) { amaxU[n] = 0u; denom[n] = 0.f; }
}

// ---- per-edge logit + segment max -----------------------------------------
__global__ __launch_bounds__(256) void k_edge_max(const int* __restrict__ src,
                                                  const int* __restrict__ dst,
                                                  const float* __restrict__ aI,
                                                  const float* __restrict__ aJ,
                                                  float* __restrict__ alphaE,
                                                  unsigned* __restrict__ amaxU,
                                                  int E) {
    int e = blockIdx.x * blockDim.x + threadIdx.x;
    if (e >= E) return;
    int s = src[e], d = dst[e];
    float al = aI[d] + aJ[s];
    al = (al >= 0.f) ? al : NEG_SLOPE * al;  // leaky_relu
    alphaE[e] = al;
    atomicMax(&amaxU[d], f2ord(al));
}

// ---- per-edge exp + segment sum -------------------------------------------
__global__ __launch_bounds__(256) void k_edge_exp(const int* __restrict__ dst,
                                                  float* __restrict__ alphaE,
                                                  const unsigned* __restrict__ amaxU,
                                                  float* __restrict__ denom, int E) {
    int e = blockIdx.x * blockDim.x + threadIdx.x;
    if (e >= E) return;
    int d = dst[e];
    float ev = expf(alphaE[e] - ord2f(amaxU[d]));
    alphaE[e] = ev;
    atomicAdd(&denom[d], ev);
}

// ---- message scatter: one wave per edge, 2 features/lane ------------------
__global__ __launch_bounds__(256) void k_scatter(const int* __restrict__ src,
                                                 const int* __restrict__ dst,
                                                 const float* __restrict__ H,
                                                 const float* __restrict__ alphaE,
                                                 const float* __restrict__ denom,
                                                 float* __restrict__ agg, int E) {
    int gid  = blockIdx.x * blockDim.x + threadIdx.x;
    int e    = gid >> 5;
    if (e >= E) return;
    int lane = gid & 31;
    int s = src[e], d = dst[e];
    float coeff = alphaE[e] / (denom[d] + 1e-16f);
    const float2 hv = *(const float2*)(H + (size_t)s * FDIM + lane * 2);
    float* o = agg + (size_t)d * FDIM + lane * 2;
    atomicAdd(o,     coeff * hv.x);
    atomicAdd(o + 1, coeff * hv.y);
}

// ---- fused output projection (64->40) + log_softmax -----------------------
#define OUTC 40
__global__ __launch_bounds__(256) void k_mlp_out(const float* __restrict__ H3,
                                                 const float* __restrict__ Wp2,
                                                 const float* __restrict__ bp2,
                                                 float* __restrict__ out, int M) {
    __shared__ float sW[OUTC * FDIM];
    __shared__ float sB[OUTC];
    for (int i = threadIdx.x; i < OUTC * FDIM; i += blockDim.x) sW[i] = Wp2[i];
    if (threadIdx.x < OUTC) sB[threadIdx.x] = bp2[threadIdx.x];
    __syncthreads();

    int n = blockIdx.x * blockDim.x + threadIdx.x;
    if (n >= M) return;

    float4 hreg[16];
    const float4* hp = (const float4*)(H3 + (size_t)n * FDIM);
#pragma unroll
    for (int i = 0; i < 16; ++i) hreg[i] = hp[i];

    float logits[OUTC];
    float mx = -3.4e38f;
    for (int c = 0; c < OUTC; ++c) {
        const float* w = &sW[c * FDIM];
        float s = sB[c];
#pragma unroll
        for (int i = 0; i < 16; ++i) {
            float4 h4 = hreg[i];
            s = fmaf(h4.x, w[4 * i + 0], s);
            s = fmaf(h4.y, w[4 * i + 1], s);
            s = fmaf(h4.z, w[4 * i + 2], s);
            s = fmaf(h4.w, w[4 * i + 3], s);
        }
        logits[c] = s;
        mx = fmaxf(mx, s);
    }
    float se = 0.f;
    for (int c = 0; c < OUTC; ++c) se += expf(logits[c] - mx);
    float lse = mx + logf(se);
    float* o = out + (size_t)n * OUTC;
    for (int c = 0; c < OUTC; ++c) o[c] = logits[c] - lse;
}

// ---------------------------------------------------------------------------
// Host orchestration
// ---------------------------------------------------------------------------
extern "C" void kernel_launch(void* const* d_in, const int* in_sizes, int n_in,
                              void* d_out, int out_size, void* d_ws, size_t ws_size,
                              hipStream_t stream) {
    const float* x    = (const float*)d_in[0];
    const int*   ei   = (const int*)d_in[1];
    const float* W1   = (const float*)d_in[2];
    const float* b1   = (const float*)d_in[3];
    const float* att1 = (const float*)d_in[4];
    const float* bi1  = (const float*)d_in[5];
    const float* W2   = (const float*)d_in[6];
    const float* b2   = (const float*)d_in[7];
    const float* att2 = (const float*)d_in[8];
    const float* bi2  = (const float*)d_in[9];
    const float* Wp1  = (const float*)d_in[10];
    const float* bp1  = (const float*)d_in[11];
    const float* Wp2  = (const float*)d_in[12];
    const float* bp2  = (const float*)d_in[13];

    const int N = in_sizes[0] / FDIM;   // 50000
    const int E = in_sizes[1] / 2;      // 800000
    const int* src = ei;
    const int* dst = ei + E;

    // workspace carve-up (floats)
    float*    H      = (float*)d_ws;                       // N*64
    float*    AGG    = H + (size_t)N * FDIM;               // N*64
    float*    aI     = AGG + (size_t)N * FDIM;             // N
    float*    aJ     = aI + N;                             // N
    unsigned* amaxU  = (unsigned*)(aJ + N);                // N
    float*    denom  = (float*)(amaxU + N);                // N
    float*    alphaE = denom + N;                          // E

    const int TB = 256;
    const dim3 gGemm(N / 16), bGemm(128);
    const int gNode  = (N + TB - 1) / TB;
    const int gEdge  = (E + TB - 1) / TB;
    const int gElem  = (N * FDIM + TB - 1) / TB;
    const int gScat  = ((E * 32) + TB - 1) / TB;

    // ---- GAT layer 1 ------------------------------------------------------
    k_gemm64<false><<<gGemm, bGemm, 0, stream>>>(x, W1, b1, H, N);
    k_att<<<gNode, TB, 0, stream>>>(H, att1, aI, aJ, N);
    k_init_agg<<<gElem, TB, 0, stream>>>(AGG, bi1, N * FDIM);
    k_init_nd<<<gNode, TB, 0, stream>>>(amaxU, denom, N);
    k_edge_max<<<gEdge, TB, 0, stream>>>(src, dst, aI, aJ, alphaE, amaxU, E);
    k_edge_exp<<<gEdge, TB, 0, stream>>>(dst, alphaE, amaxU, denom, E);
    k_scatter<<<gScat, TB, 0, stream>>>(src, dst, H, alphaE, denom, AGG, E);
    // AGG now holds pre-relu layer-1 output; relu folded into next GEMM load.

    // ---- GAT layer 2 ------------------------------------------------------
    k_gemm64<true><<<gGemm, bGemm, 0, stream>>>(AGG, W2, b2, H, N);
    k_att<<<gNode, TB, 0, stream>>>(H, att2, aI, aJ, N);
    k_init_agg<<<gElem, TB, 0, stream>>>(AGG, bi2, N * FDIM);
    k_init_nd<<<gNode, TB, 0, stream>>>(amaxU, denom, N);
    k_edge_max<<<gEdge, TB, 0, stream>>>(src, dst, aI, aJ, alphaE, amaxU, E);
    k_edge_exp<<<gEdge, TB, 0, stream>>>(dst, alphaE, amaxU, denom, E);
    k_scatter<<<gScat, TB, 0, stream>>>(src, dst, H, alphaE, denom, AGG, E);

    // ---- post-MP MLP + log_softmax ---------------------------------------
    k_gemm64<true><<<gGemm, bGemm, 0, stream>>>(AGG, Wp1, bp1, H, N);
    k_mlp_out<<<gNode, TB, 0, stream>>>(H, Wp2, bp2, (float*)d_out, N);
}